// proposal_layer_32169305047373
// MI455X (gfx1250) — compile-verified
//
#include <hip/hip_runtime.h>
#include <math.h>
#include <stdint.h>

typedef __attribute__((ext_vector_type(2))) float v2f;
typedef __attribute__((ext_vector_type(8))) float v8f;
typedef __attribute__((vector_size(16)))    int   v4i;

constexpr int Hd = 128, Wd = 128, CIN = 1024;
constexpr int NPIX  = Hd * Wd;          // 16384
constexpr int NA    = 9;
constexpr int NANCH = NPIX * NA;        // 147456
constexpr int NCOL  = 64;               // 18 cls + 36 bbox, padded to 64
constexpr int NSORT = 262144;           // next pow2 >= NANCH
constexpr int KC    = 64;               // K-chunk staged through LDS
constexpr int NCH   = CIN / KC;         // 16 chunks
constexpr float IOU_THR = 0.7f;

// --------------------------------------------- CDNA5 async-copy primitives
typedef __attribute__((address_space(1))) v4i gv4i;   // global (device) AS
typedef __attribute__((address_space(3))) v4i lv4i;   // LDS (shared) AS

__device__ __forceinline__ void cp_async16(const float* gp, float* lp) {
#if __has_builtin(__builtin_amdgcn_global_load_async_to_lds_b128)
    __builtin_amdgcn_global_load_async_to_lds_b128(
        (gv4i*)gp, (lv4i*)lp, 0, 0);
#else
    *(float4*)lp = *(const float4*)gp;          // synchronous fallback
#endif
}

__device__ __forceinline__ void wait_async_keep2() {
#if __has_builtin(__builtin_amdgcn_global_load_async_to_lds_b128)
#if __has_builtin(__builtin_amdgcn_s_wait_asynccnt)
    __builtin_amdgcn_s_wait_asynccnt(2);
#else
    asm volatile("s_wait_asynccnt 2" ::: "memory");
#endif
#endif
}

__device__ __forceinline__ void wait_async_all() {
#if __has_builtin(__builtin_amdgcn_global_load_async_to_lds_b128)
#if __has_builtin(__builtin_amdgcn_s_wait_asynccnt)
    __builtin_amdgcn_s_wait_asynccnt(0);
#else
    asm volatile("s_wait_asynccnt 0" ::: "memory");
#endif
#endif
}

// ---------------------------------------------------------------- utilities
__global__ void zero_out_kernel(float* o, int n) {
    int i = blockIdx.x * blockDim.x + threadIdx.x;
    if (i < n) o[i] = 0.0f;
}

// Build column-major padded weights WtT[64][1024] and bias[64].
__global__ void prep_weights_kernel(const float* __restrict__ w_cls,
                                    const float* __restrict__ b_cls,
                                    const float* __restrict__ w_bbox,
                                    const float* __restrict__ b_bbox,
                                    float* __restrict__ WtT,
                                    float* __restrict__ bias) {
    int id = blockIdx.x * blockDim.x + threadIdx.x;
    if (id < CIN * NCOL) {
        int k = id % CIN, c = id / CIN;      // WtT[c][k]
        float v = 0.0f;
        if (c < 18)      v = w_cls[k * 18 + c];
        else if (c < 54) v = w_bbox[k * 36 + (c - 18)];
        WtT[(size_t)c * CIN + k] = v;
    }
    if (id < NCOL) {
        float v = 0.0f;
        if (id < 18)      v = b_cls[id];
        else if (id < 54) v = b_bbox[id - 18];
        bias[id] = v;
    }
}

// ------------------------------------------------- fused GEMM + decode + key
// One block = 16 pixel rows, 4 waves = 4 N-tiles of 16 columns.
// A (16 x 1024 fp32) is staged through LDS in 16x64 chunks with
// GLOBAL_LOAD_ASYNC_TO_LDS_B128 (double-buffered, s_wait_asynccnt),
// then fed to V_WMMA_F32_16X16X4_F32 (exact fp32 accumulate).
__global__ __launch_bounds__(128)
void gemm_decode_kernel(const float* __restrict__ X,
                        const float* __restrict__ WtT,
                        const float* __restrict__ bias,
                        const float* __restrict__ anchors,
                        const float* __restrict__ img_sz,
                        float* __restrict__ boxes,
                        float* __restrict__ scores,
                        unsigned long long* __restrict__ keys) {
    __shared__ float Xs[2][16][KC];     // 2 x 4KB double buffer
    __shared__ float tile[16 * NCOL];   // 4KB output tile

    const int m0   = blockIdx.x * 16;
    const int tid  = threadIdx.x;
    const int wave = tid >> 5;      // wave32
    const int lane = tid & 31;
    const int half = lane >> 4;     // 0: lanes 0-15, 1: lanes 16-31
    const int rc   = lane & 15;     // A row (M) / B column within tile
    const int n0   = wave * 16;

    // issue one 16x64-float chunk (256 x b128, 2 async instrs per wave)
    auto issue_chunk = [&](int kc, int buf) {
#pragma unroll
        for (int e = tid; e < 256; e += 128) {
            const int row = e >> 4, c4 = e & 15;
            cp_async16(X + (size_t)(m0 + row) * CIN + kc + c4 * 4,
                       &Xs[buf][row][c4 * 4]);
        }
    };

    issue_chunk(0, 0);

    // B column for this lane, column-major: contiguous in K.
    const float* Wcol = WtT + (size_t)(n0 + rc) * CIN;

    v8f acc = {0.f, 0.f, 0.f, 0.f, 0.f, 0.f, 0.f, 0.f};
    for (int ch = 0; ch < NCH; ++ch) {
        const int buf = ch & 1;
        if (ch + 1 < NCH) {
            issue_chunk((ch + 1) * KC, buf ^ 1);
            wait_async_keep2();         // chunk `ch` complete, next in flight
        } else {
            wait_async_all();
        }
        __syncthreads();

        // A 16x4 f32 layout: VGPR0 = {K0 | K2}, VGPR1 = {K1 | K3}.
        const float* Arow = &Xs[buf][rc][0];
        const int kb = ch * KC;
#pragma unroll
        for (int k = 0; k < KC; k += 4) {
            const int ka = k + 2 * half;
            v2f a, b;
            a.x = Arow[ka];
            a.y = Arow[ka + 1];
            b.x = Wcol[kb + ka];
            b.y = Wcol[kb + ka + 1];
            acc = __builtin_amdgcn_wmma_f32_16x16x4_f32(
                false, a, false, b, (short)0, acc, false, false);
        }
        __syncthreads();                // buf reusable for chunk ch+2
    }

    // C layout: VGPR v, lane -> M = v + 8*half, N = lane%16.
#pragma unroll
    for (int v = 0; v < 8; ++v)
        tile[(v + 8 * half) * NCOL + n0 + rc] = acc[v];
    __syncthreads();

    const float img0 = img_sz[0], img1 = img_sz[1];
    for (int slot = tid; slot < 16 * NA; slot += blockDim.x) {
        const int m  = slot / NA;
        const int aI = slot % NA;
        const float* row = tile + m * NCOL;

        // softmax over channel pairs; fg = prob[9+aI]
        const int ch = NA + aI, cho = ch ^ 1;
        const float lg = row[ch]  + bias[ch];
        const float lo = row[cho] + bias[cho];
        const float fg = 1.0f / (1.0f + expf(lo - lg));

        const int cb = 18 + 4 * aI;
        const float tx = row[cb + 0] + bias[cb + 0];
        const float ty = row[cb + 1] + bias[cb + 1];
        const float tw = row[cb + 2] + bias[cb + 2];
        const float th = row[cb + 3] + bias[cb + 3];

        const int i = (m0 + m) * NA + aI;
        const float a0 = anchors[i * 4 + 0], a1 = anchors[i * 4 + 1];
        const float a2 = anchors[i * 4 + 2], a3 = anchors[i * 4 + 3];
        const float wv = a3 - a1 + 1.0f, hv = a2 - a0 + 1.0f;
        const float xc = a0 + 0.5f * hv, yc = a1 + 0.5f * wv;
        const float xp = tx * hv + xc,   yp = ty * wv + yc;
        const float wp = expf(tw) * wv,  hp = expf(th) * hv;
        const float x0 = fmaxf(xp - 0.5f * hp, 0.0f);
        const float x1 = fminf(xp + 0.5f * hp, img0);
        const float y0 = fmaxf(yp - 0.5f * wp, 0.0f);
        const float y1 = fminf(yp + 0.5f * wp, img1);

        boxes[i * 4 + 0] = x0; boxes[i * 4 + 1] = y0;
        boxes[i * 4 + 2] = x1; boxes[i * 4 + 3] = y1;
        scores[i] = fg;

        // ascending u64 key == descending score, ties -> ascending index
        unsigned u = __float_as_uint(fg);
        u = (u & 0x80000000u) ? ~u : (u | 0x80000000u);
        keys[i] = ((unsigned long long)(~u) << 32) | (unsigned)i;
    }
}

__global__ void pad_keys_kernel(unsigned long long* keys) {
    int i = NANCH + blockIdx.x * blockDim.x + threadIdx.x;
    if (i < NSORT) keys[i] = ~0ULL;
}

// ----------------------------------------------------------------- sort/NMS
__global__ void bitonic_pass_kernel(unsigned long long* d, unsigned j, unsigned k) {
    unsigned i   = blockIdx.x * blockDim.x + threadIdx.x;
    unsigned ixj = i ^ j;
    if (ixj > i && i < NSORT) {
        unsigned long long a = d[i], b = d[ixj];
        const bool up = ((i & k) == 0);
        if ((a > b) == up) { d[i] = b; d[ixj] = a; }
    }
}

__global__ void gather_sorted_kernel(const unsigned long long* __restrict__ keys,
                                     const float* __restrict__ boxes,
                                     const float* __restrict__ scores,
                                     float* __restrict__ sboxes,
                                     float* __restrict__ sarea,
                                     float* __restrict__ sscores) {
    int rI = blockIdx.x * blockDim.x + threadIdx.x;
    if (rI >= NANCH) return;
    const unsigned idx = (unsigned)(keys[rI] & 0xffffffffu);
    const float x0 = boxes[idx * 4 + 0], y0 = boxes[idx * 4 + 1];
    const float x1 = boxes[idx * 4 + 2], y1 = boxes[idx * 4 + 3];
    sboxes[rI * 4 + 0] = x0; sboxes[rI * 4 + 1] = y0;
    sboxes[rI * 4 + 2] = x1; sboxes[rI * 4 + 3] = y1;
    sarea[rI]   = fmaxf(x1 - x0, 0.0f) * fmaxf(y1 - y0, 0.0f);
    sscores[rI] = scores[idx];
}

__global__ __launch_bounds__(1024)
void nms_kernel(const float* __restrict__ sboxes,
                const float* __restrict__ sarea,
                const float* __restrict__ sscores,
                float* __restrict__ out, int max_out) {
    constexpr int NW = NANCH / 32;      // 4608 words, 18KB LDS bitmask
    __shared__ unsigned sup[NW];
    __shared__ int   s_r;
    __shared__ int   s_ptr;
    __shared__ float s_box[5];
    __shared__ float s_sc;

    const int tid = threadIdx.x;
    for (int w = tid; w < NW; w += blockDim.x) sup[w] = 0u;
    if (tid == 0) s_ptr = 0;
    __syncthreads();

    float* out_scores = out + 4 * max_out;

    for (int oi = 0; oi < max_out; ++oi) {
        if (tid == 0) {
            int r = -1;
            int w = s_ptr >> 5;
            unsigned first_mask = ~0u << (s_ptr & 31);
            while (w < NW) {
                const unsigned avail = ~sup[w] & first_mask;
                if (avail) { r = (w << 5) + __builtin_ctz(avail); break; }
                first_mask = ~0u;
                ++w;
            }
            if (r >= 0) {
                sup[r >> 5] |= 1u << (r & 31);
                s_ptr = r + 1;
                s_box[0] = sboxes[r * 4 + 0]; s_box[1] = sboxes[r * 4 + 1];
                s_box[2] = sboxes[r * 4 + 2]; s_box[3] = sboxes[r * 4 + 3];
                s_box[4] = sarea[r];
                s_sc     = sscores[r];
            }
            s_r = r;
        }
        __syncthreads();
        const int r = s_r;
        if (r < 0) break;   // uniform; output tail already zero-filled

        if (tid == 0) {
            out[oi * 4 + 0] = s_box[0]; out[oi * 4 + 1] = s_box[1];
            out[oi * 4 + 2] = s_box[2]; out[oi * 4 + 3] = s_box[3];
            out_scores[oi]  = s_sc;
        }
        const float X0 = s_box[0], Y0 = s_box[1], X1 = s_box[2], Y1 = s_box[3];
        const float AR = s_box[4];

        const int w0 = (r + 1) >> 5;
        for (int w = w0 + tid; w < NW; w += blockDim.x) {
            const unsigned v = sup[w];
            if (v == ~0u) continue;
            unsigned nb = 0;
            const int base = w << 5;
            for (int b = 0; b < 32; ++b) {
                if ((v >> b) & 1u) continue;
                const int j = base + b;
                const float* bj = sboxes + (size_t)j * 4;
                const float ix0 = fmaxf(bj[0], X0);
                const float iy0 = fmaxf(bj[1], Y0);
                const float ix1 = fminf(bj[2], X1);
                const float iy1 = fminf(bj[3], Y1);
                const float inter = fmaxf(ix1 - ix0, 0.0f) * fmaxf(iy1 - iy0, 0.0f);
                const float iou   = inter / (sarea[j] + AR - inter + 1e-9f);
                if (iou > IOU_THR) nb |= (1u << b);
            }
            if (nb) sup[w] = v | nb;   // word statically owned: no atomic needed
        }
        __syncthreads();
    }
}

// ------------------------------------------------------------------- launch
extern "C" void kernel_launch(void* const* d_in, const int* in_sizes, int n_in,
                              void* d_out, int out_size, void* d_ws, size_t ws_size,
                              hipStream_t stream) {
    const float* X       = (const float*)d_in[0];
    const float* anchors = (const float*)d_in[1];
    const float* img_sz  = (const float*)d_in[2];
    const float* w_cls   = (const float*)d_in[3];
    const float* b_cls   = (const float*)d_in[4];
    const float* w_bbox  = (const float*)d_in[5];
    const float* b_bbox  = (const float*)d_in[6];
    float* out = (float*)d_out;
    const int max_out = out_size / 5;   // (max_out,4) proposals + (max_out) scores

    auto al = [](size_t x) { return (x + 255) & ~(size_t)255; };
    char* p = (char*)d_ws;
    float* WtT     = (float*)p;               p += al((size_t)CIN * NCOL * 4);
    float* bias    = (float*)p;               p += al((size_t)NCOL * 4);
    float* boxes   = (float*)p;               p += al((size_t)NANCH * 4 * 4);
    float* scores  = (float*)p;               p += al((size_t)NANCH * 4);
    unsigned long long* keys = (unsigned long long*)p; p += al((size_t)NSORT * 8);
    float* sboxes  = (float*)p;               p += al((size_t)NANCH * 4 * 4);
    float* sarea   = (float*)p;               p += al((size_t)NANCH * 4);
    float* sscores = (float*)p;               p += al((size_t)NANCH * 4);
    (void)ws_size; (void)in_sizes; (void)n_in;

    zero_out_kernel<<<(out_size + 255) / 256, 256, 0, stream>>>(out, out_size);
    prep_weights_kernel<<<(CIN * NCOL + 255) / 256, 256, 0, stream>>>(
        w_cls, b_cls, w_bbox, b_bbox, WtT, bias);
    gemm_decode_kernel<<<NPIX / 16, 128, 0, stream>>>(
        X, WtT, bias, anchors, img_sz, boxes, scores, keys);
    pad_keys_kernel<<<(NSORT - NANCH + 255) / 256, 256, 0, stream>>>(keys);

    for (unsigned k = 2; k <= (unsigned)NSORT; k <<= 1)
        for (unsigned j = k >> 1; j > 0; j >>= 1)
            bitonic_pass_kernel<<<NSORT / 256, 256, 0, stream>>>(keys, j, k);

    gather_sorted_kernel<<<(NANCH + 255) / 256, 256, 0, stream>>>(
        keys, boxes, scores, sboxes, sarea, sscores);
    nms_kernel<<<1, 1024, 0, stream>>>(sboxes, sarea, sscores, out, max_out);
}